// FullChain_58815282152074
// MI455X (gfx1250) — compile-verified
//
#include <hip/hip_runtime.h>

typedef float v2f __attribute__((ext_vector_type(2)));
typedef float v8f __attribute__((ext_vector_type(8)));

// ---------------------------------------------------------------------------
// Kernel 1: fold W2/W3/biases into the bilinear weights.
//   ws[0 .. 4095]    = W_bil_eff[o][d][e]  (o*256 + d*16 + e)
//   ws[4096 .. 4111] = bias_eff[o]
// ---------------------------------------------------------------------------
__global__ __launch_bounds__(256) void fold_weights_kernel(
    const float* __restrict__ W_bil,  // [k][d][e] = [16][16][16]
    const float* __restrict__ b_bil,  // [16]
    const float* __restrict__ W2,     // [o][k]
    const float* __restrict__ b2,     // [16]
    const float* __restrict__ W3,     // [o][k]
    const float* __restrict__ b3,     // [16]
    float* __restrict__ ws)
{
    __shared__ float weff[256];       // W_eff[p][k] = sum_o W3[p][o]*W2[o][k]
    const int t = threadIdx.x;
    {
        const int p = t >> 4, k = t & 15;
        float acc = 0.f;
#pragma unroll
        for (int o = 0; o < 16; ++o) acc += W3[p * 16 + o] * W2[o * 16 + k];
        weff[t] = acc;
    }
    __syncthreads();

    // W_bil_eff[p][d][e] = sum_k W_eff[p][k] * W_bil[k][d][e]
    for (int idx = t; idx < 4096; idx += 256) {
        const int p = idx >> 8, de = idx & 255;
        float acc = 0.f;
#pragma unroll
        for (int k = 0; k < 16; ++k) acc += weff[p * 16 + k] * W_bil[k * 256 + de];
        ws[idx] = acc;
    }

    if (t < 16) {
        float acc = b3[t];
#pragma unroll
        for (int k = 0; k < 16; ++k)
            acc += weff[t * 16 + k] * b_bil[k] + W3[t * 16 + k] * b2[k];
        ws[4096 + t] = acc;
    }
}

// ---------------------------------------------------------------------------
// Kernel 2: one block per node i (n = b*M + i). 8 waves per block.
//   T_i[o][e] = sum_d x_i[d] * W_bil_eff[o][d][e]      (block -> LDS)
//   For each 16-row j-tile:  D[j][o] = Xj[16x16] * T_i^T[16x16] + bias_eff
//   via 4 chained V_WMMA_F32_16X16X4_F32, then self-edge-compacted NT stores.
//   Compaction is tile-uniform except in the single tile containing i, so
//   63/64 tiles use branch-free stores with immediate offsets.
// ---------------------------------------------------------------------------
__global__ __launch_bounds__(256) void edge_kernel(
    const float* __restrict__ nodes,  // [N][16]
    const float* __restrict__ ws,     // folded weights + bias
    float* __restrict__ out,          // [B*M*(M-1)][16]
    const int* __restrict__ num_batches,
    int N)
{
    __shared__ float ldsX[16];        // x_i
    __shared__ float ldsT[256];       // T_i[o][e]

    const int M = N / num_batches[0];
    const int n = blockIdx.x;         // global node index = b*M + i
    const int i = n % M;              // node within its batch
    const int t = threadIdx.x;

    if (t < 16) ldsX[t] = nodes[n * 16 + t];
    __syncthreads();

    {   // T_i[o][e]
        const int o = t >> 4, e = t & 15;
        const float* wp = ws + o * 256 + e;
        float acc = 0.f;
#pragma unroll
        for (int d = 0; d < 16; ++d) acc += ldsX[d] * wp[d * 16];
        ldsT[o * 16 + e] = acc;
    }
    __syncthreads();

    const int wave    = t >> 5;
    const int lane    = t & 31;
    const int halfsel = lane >> 4;    // 0: lanes 0-15, 1: lanes 16-31
    const int lcol    = lane & 15;

    // B operand (T_i^T): B[K=e][N=o] = T_i[o][e].
    // Per K-step s, this lane holds K = 4s + 2*halfsel and K+1, column N = lcol.
    v2f bv[4];
#pragma unroll
    for (int s = 0; s < 4; ++s) {
        const int e0 = 4 * s + 2 * halfsel;
        bv[s].x = ldsT[lcol * 16 + e0];
        bv[s].y = ldsT[lcol * 16 + e0 + 1];
    }

    // Accumulator seeded with bias_eff[o]; C/D element c[r] = (row r+8*halfsel, col lcol)
    const float bias = ws[4096 + lcol];
    v8f cinit;
#pragma unroll
    for (int r = 0; r < 8; ++r) cinit[r] = bias;

    const float* Xb    = nodes + (size_t)(n - i) * 16;       // batch base
    float*       obase = out + (size_t)n * (size_t)(M - 1) * 16;
    const int ntiles = M >> 4;
    const int itile  = i >> 4;        // the only tile with a dropped row

    for (int jt = wave; jt < ntiles; jt += 8) {
        // A operand: row M = lcol of the j-tile; per step s lane holds
        // K = 4s + 2*halfsel and K+1  ->  one b64 load per step.
        const float* xrow = Xb + (size_t)(jt * 16 + lcol) * 16;
        v2f av[4];
#pragma unroll
        for (int s = 0; s < 4; ++s)
            av[s] = *(const v2f*)(xrow + 4 * s + 2 * halfsel);

        v8f c = cinit;
        c = __builtin_amdgcn_wmma_f32_16x16x4_f32(false, av[0], false, bv[0], (short)0, c, false, false);
        c = __builtin_amdgcn_wmma_f32_16x16x4_f32(false, av[1], false, bv[1], (short)0, c, false, false);
        c = __builtin_amdgcn_wmma_f32_16x16x4_f32(false, av[2], false, bv[2], (short)0, c, false, false);
        c = __builtin_amdgcn_wmma_f32_16x16x4_f32(false, av[3], false, bv[3], (short)0, c, false, false);

        if (jt != itile) {
            // Tile-uniform compaction: row = j - (jt > itile).
            // One address computation; 8 stores at immediate offsets r*64B.
            const int row0 = jt * 16 + 8 * halfsel - (jt > itile);
            float* p = obase + (size_t)row0 * 16 + lcol;
#pragma unroll
            for (int r = 0; r < 8; ++r)
                __builtin_nontemporal_store(c[r], p + (size_t)r * 16);
        } else {
            // The single tile containing i: per-row predication + shift.
            const int jbase = jt * 16;
#pragma unroll
            for (int r = 0; r < 8; ++r) {
                const int j = jbase + r + 8 * halfsel;
                if (j != i) {
                    const int row = j - (j > i);
                    __builtin_nontemporal_store(c[r], obase + (size_t)row * 16 + lcol);
                }
            }
        }
    }
}

// ---------------------------------------------------------------------------
extern "C" void kernel_launch(void* const* d_in, const int* in_sizes, int n_in,
                              void* d_out, int out_size, void* d_ws, size_t ws_size,
                              hipStream_t stream) {
    const float* nodes = (const float*)d_in[0];
    const float* W_bil = (const float*)d_in[1];
    const float* b_bil = (const float*)d_in[2];
    const float* W2    = (const float*)d_in[3];
    const float* b2    = (const float*)d_in[4];
    const float* W3    = (const float*)d_in[5];
    const float* b3    = (const float*)d_in[6];
    const int*   nb    = (const int*)d_in[7];
    float* ws  = (float*)d_ws;
    float* out = (float*)d_out;

    const int N = in_sizes[0] / 16;   // 4096 nodes

    fold_weights_kernel<<<1, 256, 0, stream>>>(W_bil, b_bil, W2, b2, W3, b3, ws);
    edge_kernel<<<N, 256, 0, stream>>>(nodes, ws, out, nb, N);
}